// Generator_53180285059230
// MI455X (gfx1250) — compile-verified
//
#include <hip/hip_runtime.h>
#include <math.h>

typedef __attribute__((ext_vector_type(2))) float v2f;
typedef __attribute__((ext_vector_type(8))) float v8f;

#define IMG_H 2048
#define IMG_W 4096
#define PH 256
#define PW 256
#define PI_F 3.14159265358979323846f

// ---------------------------------------------------------------------------
// Kernel 1: per-pixel precompute (batch-invariant part).
//   e = (Ry(pitch)*Rx(roll)) * [ (x-cx)/f, (y-cy)/f, 1 ]   via V_WMMA_F32_16X16X4_F32
//   phi   = asin(e_z/|e|)              -> uj -> {v0, v1, dv}   (yaw-invariant)
//   theta0= atan2(e_y, e_x)            -> base_u               (yaw adds a pure shift)
// One WMMA per wave handles 16 pixels (A = 3x3 rotation padded to 16x4,
// B = 4x16 of pixel ray columns, D rows 0..2 = e_x/e_y/e_z of pixel N=lane).
// ---------------------------------------------------------------------------
__global__ __launch_bounds__(256) void equi_precompute_wmma(
    const float* __restrict__ fov, const float* __restrict__ roll,
    const float* __restrict__ pitch, float4* __restrict__ ws)
{
  const int tid  = blockIdx.x * 256 + threadIdx.x;
  const int lane = threadIdx.x & 31;          // wave32
  const int gw   = tid >> 5;                  // global wave id, 0..4095
  const int m    = lane & 15;                 // A row / B column (pixel-in-group)
  const int pix  = gw * 16 + m;               // 0..65535
  const int xi   = pix & (PW - 1);
  const int yi   = pix >> 8;
  const bool lo  = lane < 16;                 // VGPRs hold K=0,1 (lo) or K=2,3 (hi)

  const float fv = fov[0];
  const float rl = roll[0];
  const float pt = pitch[0];
  const float sr = sinf(rl), cr = cosf(rl);
  const float sp = sinf(pt), cp = cosf(pt);
  const float f  = (float)PW / (2.0f * tanf(fv * (PI_F / 180.0f) * 0.5f));
  const float cx = (float)PW * 0.5f;
  const float cy = (float)PH * 0.5f;

  const float px = ((float)xi - cx) / f;
  const float py = ((float)yi - cy) / f;

  // M3 = Ry*Rx:
  //  row0: [ cp,    sp*sr,  sp*cr ]
  //  row1: [ 0,     cr,    -sr    ]
  //  row2: [-sp,    cp*sr,  cp*cr ]
  // A-matrix 16x4 f32 layout: VGPR0 = A[m][lo?0:2], VGPR1 = A[m][lo?1:3]
  float a0 = lo ? ((m == 0) ? cp      : (m == 1) ? 0.0f : (m == 2) ? -sp     : 0.0f)
                : ((m == 0) ? sp * cr : (m == 1) ? -sr  : (m == 2) ? cp * cr : 0.0f);
  float a1 = lo ? ((m == 0) ? sp * sr : (m == 1) ? cr   : (m == 2) ? cp * sr : 0.0f)
                : 0.0f;
  // B-matrix 4x16 f32 layout: VGPR0 = B[lo?0:2][m], VGPR1 = B[lo?1:3][m]
  // column m of B = [px, py, 1, 0]^T for pixel m
  float b0 = lo ? px : 1.0f;   // K=0 -> px ; K=2 -> 1
  float b1 = lo ? py : 0.0f;   // K=1 -> py ; K=3 -> 0

  v2f A; A[0] = a0; A[1] = a1;
  v2f B; B[0] = b0; B[1] = b1;
  v8f C = {};
  // (neg_a, A, neg_b, B, c_mod, C, reuse_a, reuse_b)
  v8f D = __builtin_amdgcn_wmma_f32_16x16x4_f32(
      false, A, false, B, (short)0, C, false, false);

  // Lanes 0..15 hold D[row][N=lane]: row0=e_x, row1=e_y, row2=e_z
  if (lo) {
    const float ex = D[0];
    const float ey = D[1];
    const float ez = D[2];
    const float nrm  = sqrtf(ex * ex + ey * ey + ez * ez);
    const float phi  = asinf(ez / nrm);
    const float th0  = atan2f(ey, ex);

    const float base_u = (th0 - PI_F) * ((float)IMG_W / (2.0f * PI_F)) + 0.5f;

    float vj  = (phi - 0.5f * PI_F) * ((float)IMG_H / PI_F) + 0.5f;
    float ujm = fmodf(vj, (float)IMG_H);
    if (ujm < 0.0f) ujm += (float)IMG_H;
    const float v0f = floorf(ujm);
    const float dv  = ujm - v0f;
    int v0 = (int)v0f;
    v0 = v0 < 0 ? 0 : (v0 > IMG_H - 1 ? IMG_H - 1 : v0);
    int v1 = v0 + 1;
    if (v1 > IMG_H - 1) v1 = IMG_H - 1;

    float4 o;
    o.x = base_u;
    o.y = dv;
    o.z = __int_as_float(v0);
    o.w = __int_as_float(v1);
    ws[pix] = o;
  }
}

// ---------------------------------------------------------------------------
// Kernel 2: per (batch, pixel) bilinear gather + tanh.
// ui = base_u + yaw*W/(2*pi); u-wrap via &(W-1) (W is a power of two, matches
// jnp.mod incl. negatives); v indices precomputed (yaw-invariant).
// Image (100 MB) is L2-resident on MI455X (192 MB L2), so the 12 gathers per
// pixel hit L2. Stores are fully coalesced in x.
// ---------------------------------------------------------------------------
__global__ __launch_bounds__(256) void equi_sample_tanh(
    const float* __restrict__ img, const float* __restrict__ yaw,
    const float4* __restrict__ ws, float* __restrict__ out)
{
  const int b   = blockIdx.x >> 8;                       // batch 0..63 (uniform)
  const int pix = ((blockIdx.x & 255) << 8) | threadIdx.x; // 0..65535, x-fastest

  const float shift = yaw[b] * ((float)IMG_W / (2.0f * PI_F));

  const float4 t = ws[pix];
  const float ui  = t.x + shift;
  const float u0f = floorf(ui);
  const float du  = ui - u0f;
  const int u0 = ((int)u0f) & (IMG_W - 1);
  const int u1 = (u0 + 1) & (IMG_W - 1);
  const float dv = t.y;
  const int v0 = __float_as_int(t.z);
  const int v1 = __float_as_int(t.w);

  const float w00 = (1.0f - du) * (1.0f - dv);
  const float w01 = du * (1.0f - dv);
  const float w10 = (1.0f - du) * dv;
  const float w11 = du * dv;

  const int r0 = v0 * IMG_W;
  const int r1 = v1 * IMG_W;
  const int i00 = r0 + u0, i01 = r0 + u1;
  const int i10 = r1 + u0, i11 = r1 + u1;

#pragma unroll
  for (int c = 0; c < 3; ++c) {
    const float* __restrict__ ch = img + (size_t)c * ((size_t)IMG_H * IMG_W);
    const float acc = ch[i00] * w00 + ch[i01] * w01 +
                      ch[i10] * w10 + ch[i11] * w11;
    // out layout: (B, C, PH, PW), pix = y*256 + x
    out[(((size_t)(b * 3 + c)) << 16) | (size_t)pix] = tanhf(acc);
  }
}

extern "C" void kernel_launch(void* const* d_in, const int* in_sizes, int n_in,
                              void* d_out, int out_size, void* d_ws, size_t ws_size,
                              hipStream_t stream) {
  const float* img   = (const float*)d_in[0]; // 3 x 2048 x 4096
  const float* fov   = (const float*)d_in[1]; // 1
  const float* roll  = (const float*)d_in[2]; // 1
  const float* pitch = (const float*)d_in[3]; // 1
  const float* yaw   = (const float*)d_in[4]; // 64
  float4* ws  = (float4*)d_ws;                // 65536 * 16 B = 1 MB
  float*  out = (float*)d_out;                // 64*3*256*256

  // 65536 pixels / 16 per wave-WMMA = 4096 waves = 512 blocks x 256 threads
  equi_precompute_wmma<<<512, 256, 0, stream>>>(fov, roll, pitch, ws);

  // 64 batches x 256 y-rows of blocks, 256 threads (x) each
  equi_sample_tanh<<<64 * 256, 256, 0, stream>>>(img, yaw, ws, out);
}